// RPNProposal_52544629899845
// MI455X (gfx1250) — compile-verified
//
#include <hip/hip_runtime.h>
#include <math.h>

// ---------------- constants ----------------
#define PRE_NMS   6000
#define POST_NMS  300
#define IOU_THR   0.7f
#define WORDS     188          // ceil(6000/32)
#define HWPR      376          // uint16 halfwords per supp row (2*WORDS, last one padding)
#define CAND_CAP  8192
#define NBINS     2048
#define TILES     375          // 6000/16

typedef __attribute__((ext_vector_type(2))) float v2f;
typedef __attribute__((ext_vector_type(8))) float v8f;

// ---------------- helpers ----------------
__device__ __forceinline__ unsigned f2k(float f) {
    unsigned u = __float_as_uint(f);
    return (u & 0x80000000u) ? ~u : (u | 0x80000000u);
}

// decode + clip exactly as reference (means=0, stds=1)
__device__ __forceinline__ void decode_clip(const float4 a, const float4 d,
                                            float W, float H, float S,
                                            float4* box, bool* valid) {
    float w  = a.z - a.x + 1.0f;
    float h  = a.w - a.y + 1.0f;
    float cx = a.x + 0.5f * w;
    float cy = a.y + 0.5f * h;
    float pcx = d.x * w + cx;
    float pcy = d.y * h + cy;
    float pw  = expf(d.z) * w;
    float ph  = expf(d.w) * h;
    float x1 = pcx - 0.5f * pw;
    float y1 = pcy - 0.5f * ph;
    float x2 = pcx + 0.5f * pw - 1.0f;
    float y2 = pcy + 0.5f * ph - 1.0f;
    x1 = fminf(fmaxf(x1, 0.0f), W - 1.0f);
    y1 = fminf(fmaxf(y1, 0.0f), H - 1.0f);
    x2 = fminf(fmaxf(x2, 0.0f), W - 1.0f);
    y2 = fminf(fmaxf(y2, 0.0f), H - 1.0f);
    box->x = x1; box->y = y1; box->z = x2; box->w = y2;
    *valid = ((x2 - x1 + 1.0f) >= S) && ((y2 - y1 + 1.0f) >= S);
}

// ---------------- K0: zero histograms/ctrl ----------------
__global__ void k_init(unsigned* hist1, unsigned* hist2, unsigned* ctrl) {
    int t = blockIdx.x * blockDim.x + threadIdx.x;
    if (t < NBINS) { hist1[t] = 0u; hist2[t] = 0u; }
    if (t < 64)    ctrl[t] = 0u;
}

// ---------------- K1: per-anchor masked-score sort key ----------------
__global__ void k_keys(const float* __restrict__ deltas, const float* __restrict__ anchors,
                       const float* __restrict__ scores,
                       const int* pH, const int* pW, const int* pS,
                       unsigned* __restrict__ keys, int N) {
    int i = blockIdx.x * blockDim.x + threadIdx.x;
    if (i >= N) return;
    float4 a = ((const float4*)anchors)[i];
    float4 d = ((const float4*)deltas)[i];
    float4 box; bool valid;
    decode_clip(a, d, (float)*pW, (float)*pH, (float)*pS, &box, &valid);
    float s = valid ? scores[i] : -INFINITY;
    keys[i] = f2k(s);
}

// ---------------- K2: coarse histogram (key>>21) ----------------
__global__ void k_hist1(const unsigned* __restrict__ keys, unsigned* __restrict__ hist, int N) {
    __shared__ unsigned lh[NBINS];
    for (int b = threadIdx.x; b < NBINS; b += blockDim.x) lh[b] = 0u;
    __syncthreads();
    for (int i = blockIdx.x * blockDim.x + threadIdx.x; i < N; i += gridDim.x * blockDim.x)
        atomicAdd(&lh[keys[i] >> 21], 1u);
    __syncthreads();
    for (int b = threadIdx.x; b < NBINS; b += blockDim.x)
        if (lh[b]) atomicAdd(&hist[b], lh[b]);
}

// ---------------- K3: find coarse bin ----------------
__global__ void k_sel1(const unsigned* __restrict__ hist, unsigned* ctrl) {
    unsigned cum = 0;
    for (int b = NBINS - 1; b >= 0; --b) {
        unsigned c = hist[b];
        cum += c;
        if (cum >= PRE_NMS) { ctrl[0] = (unsigned)b; ctrl[1] = cum - c; return; }
    }
    ctrl[0] = 0u; ctrl[1] = 0u;
}

// ---------------- K4: fine histogram within coarse bin ----------------
__global__ void k_hist2(const unsigned* __restrict__ keys, const unsigned* __restrict__ ctrl,
                        unsigned* __restrict__ hist, int N) {
    __shared__ unsigned lh[NBINS];
    unsigned cb = ctrl[0];
    for (int b = threadIdx.x; b < NBINS; b += blockDim.x) lh[b] = 0u;
    __syncthreads();
    for (int i = blockIdx.x * blockDim.x + threadIdx.x; i < N; i += gridDim.x * blockDim.x) {
        unsigned k = keys[i];
        if ((k >> 21) == cb) atomicAdd(&lh[(k >> 10) & (NBINS - 1)], 1u);
    }
    __syncthreads();
    for (int b = threadIdx.x; b < NBINS; b += blockDim.x)
        if (lh[b]) atomicAdd(&hist[b], lh[b]);
}

// ---------------- K5: 22-bit threshold ----------------
__global__ void k_sel2(const unsigned* __restrict__ hist, unsigned* ctrl) {
    unsigned cum = ctrl[1], cb = ctrl[0];
    for (int f = NBINS - 1; f >= 0; --f) {
        cum += hist[f];
        if (cum >= PRE_NMS) { ctrl[2] = (cb << 11) | (unsigned)f; return; }
    }
    ctrl[2] = cb << 11;
}

// ---------------- K6: compact candidates >= threshold ----------------
__global__ void k_compact(const unsigned* __restrict__ keys, unsigned* ctrl,
                          unsigned long long* __restrict__ cand, int N) {
    int i = blockIdx.x * blockDim.x + threadIdx.x;
    if (i >= N) return;
    unsigned T = ctrl[2];
    unsigned k = keys[i];
    if ((k >> 10) >= T) {
        unsigned pos = atomicAdd(&ctrl[3], 1u);
        if (pos < CAND_CAP)
            cand[pos] = ((unsigned long long)k << 32) | (unsigned)(~(unsigned)i);
    }
}

// ---------------- K7: bitonic sort candidates in LDS, emit top-6000 indices ----------------
__global__ __launch_bounds__(512) void k_sort(const unsigned* __restrict__ ctrl,
                                              const unsigned long long* __restrict__ cand,
                                              unsigned* __restrict__ topIdx, int N) {
    __shared__ unsigned long long s[CAND_CAP];   // 64 KB of the 320 KB WGP LDS
    const int TH = 512;
    int tid = threadIdx.x;
    unsigned cc = ctrl[3]; if (cc > CAND_CAP) cc = CAND_CAP;
    for (int t = tid; t < CAND_CAP; t += TH) s[t] = (t < (int)cc) ? cand[t] : 0ull;
    __syncthreads();
    for (unsigned k = 2; k <= CAND_CAP; k <<= 1) {
        for (unsigned j = k >> 1; j > 0; j >>= 1) {
            for (unsigned t = tid; t < CAND_CAP; t += TH) {
                unsigned x = t ^ j;
                if (x > t) {
                    unsigned long long a = s[t], b = s[x];
                    bool desc = ((t & k) == 0);
                    if (desc ? (a < b) : (a > b)) { s[t] = b; s[x] = a; }
                }
            }
            __syncthreads();
        }
    }
    for (int t = tid; t < PRE_NMS; t += TH) {
        unsigned idx = ~(unsigned)s[t];
        if (idx >= (unsigned)N) idx = 0;
        topIdx[t] = idx;
    }
}

// ---------------- K8: gather + re-decode the 6000 boxes, compute areas ----------------
__global__ void k_gather(const float* __restrict__ deltas, const float* __restrict__ anchors,
                         const int* pH, const int* pW, const int* pS,
                         const unsigned* __restrict__ topIdx,
                         float* __restrict__ topBoxes, float* __restrict__ areas, int N) {
    int t = blockIdx.x * blockDim.x + threadIdx.x;
    if (t >= PRE_NMS) return;
    unsigned idx = topIdx[t];
    if (idx >= (unsigned)N) idx = 0;
    float4 a = ((const float4*)anchors)[idx];
    float4 d = ((const float4*)deltas)[idx];
    float4 box; bool valid;
    decode_clip(a, d, (float)*pW, (float)*pH, (float)*pS, &box, &valid);
    ((float4*)topBoxes)[t] = box;
    areas[t] = fmaxf(box.z - box.x, 0.0f) * fmaxf(box.w - box.y, 0.0f);
}

// ---------------- K9: suppression bit-matrix; one wave per 16x16 tile ----------------
// area_i + area_j over the tile is a rank-2 outer sum == 16x4 * 4x16 matmul:
//   A = [area_i, 1, 0, 0] rows, B = [1; area_j; 0; 0] cols -> v_wmma_f32_16x16x4_f32
__global__ __launch_bounds__(32) void k_supp(const float* __restrict__ boxes,
                                             const float* __restrict__ areas,
                                             unsigned short* __restrict__ supp16) {
    int lane = threadIdx.x;               // 0..31, full wave32
    int ti = blockIdx.y, tj = blockIdx.x;
    int i0 = ti * 16, j0 = tj * 16;
    int n = lane & 15;

    // WMMA operands. A 16x4 f32: lanes 0-15 carry K={0,1}, lanes 16-31 K={2,3},
    // both halves hold rows M=0..15 -> fill identically. B 4x16: symmetric split;
    // rows K=2,3 of B are zero so A's second-half values are don't-care.
    float ai = areas[i0 + n];
    float aj = areas[j0 + n];
    v2f A; A[0] = ai;   A[1] = 1.0f;
    v2f B; B[0] = (lane < 16) ? 1.0f : 0.0f;
           B[1] = (lane < 16) ? aj   : 0.0f;
    v8f C = {};
    v8f D = __builtin_amdgcn_wmma_f32_16x16x4_f32(false, A, false, B, (short)0, C, false, false);
    // D[v] (lane L) = area_i[v + 8*(L>=16)] + area_j[L&15]

    __shared__ float4 sbi[16];
    if (lane < 16) sbi[lane] = ((const float4*)boxes)[i0 + lane];
    __syncthreads();
    float4 bj = ((const float4*)boxes)[j0 + n];

    int mhalf = (lane >> 4) << 3;
    unsigned ball[8];
#pragma unroll
    for (int v = 0; v < 8; ++v) {
        int m = v + mhalf;
        float4 bi = sbi[m];
        float xx1 = fmaxf(bi.x, bj.x);
        float yy1 = fmaxf(bi.y, bj.y);
        float xx2 = fminf(bi.z, bj.z);
        float yy2 = fminf(bi.w, bj.w);
        float inter = fmaxf(xx2 - xx1, 0.0f) * fmaxf(yy2 - yy1, 0.0f);
        float denom = fmaxf(D[v] - inter, 1e-8f);
        bool sup = (inter > IOU_THR * denom) && ((j0 + n) > (i0 + m));
        ball[v] = (unsigned)__ballot(sup);   // bits[15:0]=row v, bits[31:16]=row v+8
    }
    if (lane < 16) {
        unsigned hw = (lane < 8) ? (ball[lane] & 0xFFFFu) : (ball[lane - 8] >> 16);
        supp16[(size_t)(i0 + lane) * HWPR + tj] = (unsigned short)hw;
    } else if (tj == TILES - 1) {
        // zero the padding halfword (bit 6000..6015 region) of each row
        supp16[(size_t)(i0 + (lane - 16)) * HWPR + (HWPR - 1)] = 0;
    }
}

// ---------------- K10: sequential greedy scan over the bit-matrix ----------------
__global__ __launch_bounds__(256) void k_nms(const unsigned* __restrict__ supp32,
                                             unsigned* __restrict__ keptList,
                                             unsigned* ctrl) {
    __shared__ unsigned remv[WORDS];
    __shared__ int scnt;
    int tid = threadIdx.x;
    if (tid < WORDS) remv[tid] = 0u;
    if (tid == 0) scnt = 0;
    __syncthreads();
    for (int i = 0; i < PRE_NMS; ++i) {
        bool kp = ((remv[i >> 5] >> (i & 31)) & 1u) == 0u;  // bit i of row i is 0 -> race-free
        if (kp) {
            if (tid < WORDS) remv[tid] |= supp32[(size_t)i * WORDS + tid];
            if (tid == 0) { keptList[scnt] = (unsigned)i; scnt++; }
        }
        __syncthreads();
        if (scnt >= POST_NMS) break;   // first 300 kept are final; later suppression irrelevant
    }
    if (tid == 0) ctrl[4] = (unsigned)(scnt < POST_NMS ? scnt : POST_NMS);
}

// ---------------- K11: write 300 output rows (zero-padded) ----------------
__global__ void k_out(const float* __restrict__ topBoxes, const unsigned* __restrict__ keptList,
                      const unsigned* __restrict__ ctrl, float* __restrict__ out) {
    int t = blockIdx.x * blockDim.x + threadIdx.x;
    if (t >= POST_NMS) return;
    unsigned kc = ctrl[4];
    float4 b = make_float4(0.0f, 0.0f, 0.0f, 0.0f);
    if ((unsigned)t < kc) b = ((const float4*)topBoxes)[keptList[t]];
    ((float4*)out)[t] = b;
}

// ---------------- launcher ----------------
extern "C" void kernel_launch(void* const* d_in, const int* in_sizes, int n_in,
                              void* d_out, int out_size, void* d_ws, size_t ws_size,
                              hipStream_t stream) {
    const float* deltas  = (const float*)d_in[0];
    const float* anchors = (const float*)d_in[1];
    const float* scores  = (const float*)d_in[2];
    const int*   pH      = (const int*)d_in[3];
    const int*   pW      = (const int*)d_in[4];
    const int*   pS      = (const int*)d_in[5];
    float* out = (float*)d_out;
    const int N = in_sizes[2];

    // workspace layout (keys region is reused for the suppression bit-matrix)
    char* base = (char*)d_ws;
    size_t KEYS_BYTES = (size_t)N * 4;                       // 15 MB
    unsigned*            keys     = (unsigned*)base;
    unsigned*            supp32   = (unsigned*)base;         // alias (keys dead by then)
    unsigned short*      supp16   = (unsigned short*)base;
    unsigned*            hist1    = (unsigned*)(base + KEYS_BYTES);
    unsigned*            hist2    = hist1 + NBINS;
    unsigned*            ctrl     = hist2 + NBINS;           // 64 u32
    unsigned long long*  cand     = (unsigned long long*)(ctrl + 64);
    unsigned*            topIdx   = (unsigned*)(cand + CAND_CAP);
    float*               topBoxes = (float*)(topIdx + PRE_NMS);
    float*               areas    = topBoxes + (size_t)PRE_NMS * 4;
    unsigned*            keptList = (unsigned*)(areas + PRE_NMS);

    const int TB = 256;
    int blocksN = (N + TB - 1) / TB;

    k_init   <<<8, TB, 0, stream>>>(hist1, hist2, ctrl);
    k_keys   <<<blocksN, TB, 0, stream>>>(deltas, anchors, scores, pH, pW, pS, keys, N);
    k_hist1  <<<1920, TB, 0, stream>>>(keys, hist1, N);
    k_sel1   <<<1, 1, 0, stream>>>(hist1, ctrl);
    k_hist2  <<<1920, TB, 0, stream>>>(keys, ctrl, hist2, N);
    k_sel2   <<<1, 1, 0, stream>>>(hist2, ctrl);
    k_compact<<<blocksN, TB, 0, stream>>>(keys, ctrl, cand, N);
    k_sort   <<<1, 512, 0, stream>>>(ctrl, cand, topIdx, N);
    k_gather <<<(PRE_NMS + TB - 1) / TB, TB, 0, stream>>>(deltas, anchors, pH, pW, pS,
                                                          topIdx, topBoxes, areas, N);
    dim3 tg(TILES, TILES);
    k_supp   <<<tg, 32, 0, stream>>>(topBoxes, areas, supp16);
    k_nms    <<<1, TB, 0, stream>>>(supp32, keptList, ctrl);
    k_out    <<<2, TB, 0, stream>>>(topBoxes, keptList, ctrl, out);
}